// VisualTokenEmbedding_18193481465967
// MI455X (gfx1250) — compile-verified
//
#include <hip/hip_runtime.h>
#include <stdint.h>

// ---------------------------------------------------------------------------
// VisualTokenEmbedding for MI455X (gfx1250), compile-only tuned.
//
// Roofline: 604MB mask read (NT streamed) + 28MB feat (L2-resident, TDM-staged
// into LDS) + 50MB out (NT stores) ~= 682MB -> ~29us @ 23.3TB/s. Pure
// bandwidth problem (~0.1 GFLOP): WMMA would add ~500x redundant FLOPs with no
// byte savings -> not used. CDNA5 paths used instead:
//   * 2x TDM tensor_load_to_lds per block (3D strided tiles -> LDS,
//     pad_enable => bank-conflict-free channel planes), s_wait_tensorcnt
//   * global_prefetch_b8 + TH_LOAD_NT on the streaming mask scan
//   * TH_STORE_NT on the streaming emb output
// ---------------------------------------------------------------------------

#define NB   4
#define MB   64
#define CCH  768
#define HF   48
#define HI   768
#define WI   768
#define NM   (NB*MB)

// workspace layout (bytes)
#define WS_BOX   0                          // float4 * 256        (box in 48-scale)
#define WS_IDX   4096                       // int4   * 256*64     (y0,x0,y1,x1)
#define WS_WTS   (4096 + 262144)            // float4 * 256*64     (bilinear weights)
#define WS_SMALL (4096 + 262144 + 262144)   // u8     * 256*2304   (masks_small)

typedef float        f32x4  __attribute__((ext_vector_type(4)));
typedef unsigned int u32x4  __attribute__((ext_vector_type(4)));
typedef int          i32x4v __attribute__((ext_vector_type(4)));
typedef int          i32x8v __attribute__((ext_vector_type(8)));

// ---------------------------------------------------------------------------
// Kernel A: one block per (n,m) mask. Scan 768x768 once (NT b128 loads +
// global_prefetch_b8). Dilated bbox == original bbox +/-8 clipped. Each set
// pixel maps to <=1 masks_small cell because window width 15 < stride 16.
// ---------------------------------------------------------------------------
__global__ __launch_bounds__(1024)
void maskScanKernel(const float* __restrict__ masks, float* out_boxes, char* ws) {
  const int nm  = blockIdx.x;
  const int tid = threadIdx.x;
  __shared__ unsigned char smask[48 * 48];
  __shared__ int sbb[4]; // ymin, ymax, xmin, xmax
  for (int i = tid; i < 48 * 48; i += 1024) smask[i] = 0;
  if (tid == 0) { sbb[0] = 1 << 30; sbb[1] = -1; sbb[2] = 1 << 30; sbb[3] = -1; }
  __syncthreads();

  const f32x4* m4 = (const f32x4*)(masks + (size_t)nm * (HI * WI));
  int ymin = 1 << 30, ymax = -1, xmin = 1 << 30, xmax = -1;
  const int nq = HI * WI / 4; // 147456 float4 per mask
  for (int q = tid; q < nq; q += 1024) {
    __builtin_prefetch(&m4[q + 4096], 0, 0);          // -> global_prefetch_b8
    f32x4 v = __builtin_nontemporal_load(&m4[q]);     // -> b128 load, TH=NT
    if (v[0] > 0.f || v[1] > 0.f || v[2] > 0.f || v[3] > 0.f) {
      int y  = q / (WI / 4);
      int xb = (q % (WI / 4)) * 4;
#pragma unroll
      for (int e = 0; e < 4; ++e) {
        if (v[e] > 0.f) {
          int x = xb + e;
          ymin = min(ymin, y); ymax = max(ymax, y);
          xmin = min(xmin, x); xmax = max(xmax, x);
          int iy = (y + 7) >> 4;
          int jx = (x + 7) >> 4;
          bool vy = (((y + 7) & 15) != 15) && (iy < 48);
          bool vx = (((x + 7) & 15) != 15) && (jx < 48);
          if (vy && vx) smask[iy * 48 + jx] = 1;
        }
      }
    }
  }
  if (ymax >= 0) {
    atomicMin(&sbb[0], ymin); atomicMax(&sbb[1], ymax);
    atomicMin(&sbb[2], xmin); atomicMax(&sbb[3], xmax);
  }
  __syncthreads();

  unsigned char* wsm = (unsigned char*)(ws + WS_SMALL) + (size_t)nm * 2304;
  for (int i = tid; i < 48 * 48; i += 1024) wsm[i] = smask[i];

  if (tid == 0) {
    float4 b48;
    if (sbb[1] < 0) {
      b48 = make_float4(0.f, 0.f, 0.f, 0.f);
    } else {
      // dilated bbox = orig +/-7 clipped, then +/-1 clipped -> +/-8 clipped; *0.0625
      b48.x = (float)max(0, sbb[2] - 8) * 0.0625f;
      b48.y = (float)max(0, sbb[0] - 8) * 0.0625f;
      b48.z = (float)min(WI - 1, sbb[3] + 8) * 0.0625f;
      b48.w = (float)min(HI - 1, sbb[1] + 8) * 0.0625f;
    }
    ((float4*)(ws + WS_BOX))[nm] = b48;
    ((float4*)out_boxes)[nm] =
        make_float4(b48.x / 48.f, b48.y / 48.f, b48.z / 48.f, b48.w / 48.f);
  }
}

// ---------------------------------------------------------------------------
// Kernel B: one block (64 threads) per (n,m). Per point: mask crop + ones
// fallback (written to d_out), bilinear tap indices + weights (valid folded
// into weights) stored to workspace.
// ---------------------------------------------------------------------------
__global__ __launch_bounds__(64)
void pointKernel(float* dout, char* ws) {
  const int nm = blockIdx.x;
  const int p  = threadIdx.x;     // 0..63
  const int i  = p >> 3, j = p & 7;
  __shared__ float scb[64];

  float4 b = ((const float4*)(ws + WS_BOX))[nm];
  const unsigned char* wsm = (const unsigned char*)(ws + WS_SMALL) + (size_t)nm * 2304;

  // nearest crop (reference _crop_mask_one)
  int x1i = max((int)floorf(b.x), 0);
  int y1i = max((int)floorf(b.y), 0);
  int x2i = min((int)floorf(b.z), 47);
  int y2i = min((int)floorf(b.w), 47);
  int h = y2i - y1i + 1;
  int w = x2i - x1i + 1;
  int yi = y1i + (i * h) / 8;
  int xi = x1i + (j * w) / 8;
  float crop = (float)wsm[yi * 48 + xi];
  scb[p] = crop;
  __syncthreads();
  float s = 0.f;
  for (int k = 0; k < 64; ++k) s += scb[k];
  float roim = (s > 0.f) ? crop : 1.0f;
  dout[1024 + nm * 64 + p] = roim;

  // bilinear geometry (reference _roi_align_one / _bilinear)
  float bw = fmaxf(b.z - b.x, 1.0f) * 0.125f;
  float bh = fmaxf(b.w - b.y, 1.0f) * 0.125f;
  float px = b.x + ((float)j + 0.5f) * bw;
  float py = b.y + ((float)i + 0.5f) * bh;
  bool valid = (py > -1.0f) && (py < 48.f) && (px > -1.0f) && (px < 48.f);
  float yc = fminf(fmaxf(py, 0.f), 47.f);
  float xc = fminf(fmaxf(px, 0.f), 47.f);
  int y0 = (int)floorf(yc);
  int x0 = (int)floorf(xc);
  int y1 = min(y0 + 1, 47);
  int x1 = min(x0 + 1, 47);
  float ly = yc - (float)y0, lx = xc - (float)x0;
  float hy = 1.f - ly, hx = 1.f - lx;
  float4 wv = make_float4(hy * hx, hy * lx, ly * hx, ly * lx);
  if (!valid) wv = make_float4(0.f, 0.f, 0.f, 0.f);
  ((int4*)(ws + WS_IDX))[nm * 64 + p]   = make_int4(y0, x0, y1, x1);
  ((float4*)(ws + WS_WTS))[nm * 64 + p] = wv;
}

// ---------------------------------------------------------------------------
// Kernel C: block = (channel-block of 4, n). Two TDM descriptors each stage a
// 2-channel 48x48 tile of feat[n] into LDS (x=48 stride 1, y=48 stride 48,
// c stride 2304) with a 1-DWORD pad per 256 DWORDs. Padded plane bases:
// {0, 2313, 4626, 6939} == {0,9,18,27} mod 64 banks -> the 4 channel-threads
// sharing one spatial tap hit 4 distinct LDS banks. Then 64 masks x 256
// outputs each: thread = (p = tid&63, c = tid>>6), NT emb stores coalesced
// over p.
// ---------------------------------------------------------------------------
#define CB 4
#define DESC_FLOATS (2 * 2304 + (2 * 2304) / 256)   // 4626 padded dwords / descriptor
#define TILE_FLOATS (2 * DESC_FLOATS + 8)

#if __has_builtin(__builtin_amdgcn_tensor_load_to_lds)
__device__ __forceinline__ void tdm_load_2ch(const float* gsrc, const float* ldsdst) {
  uint64_t gaddr   = (uint64_t)(uintptr_t)gsrc;
  uint32_t lds_off = (uint32_t)(uintptr_t)ldsdst;
  u32x4 g0;
  g0[0] = 1u;                                           // count=1, user mode
  g0[1] = lds_off;                                      // lds_addr [63:32]
  g0[2] = (uint32_t)(gaddr & 0xFFFFFFFFu);              // global_addr[31:0]
  g0[3] = (uint32_t)((gaddr >> 32) & 0x01FFFFFFu)       // global_addr[56:32]
          | 0x80000000u;                                // type=2 [127:126]
  i32x8v g1;
  g1[0] = (int)((2u << 16)        // data_size = 4B
                | (1u << 20)      // pad_enable
                | (7u << 22));    // pad_interval = 256 DWORDs (pad_amount=0 -> 1 DWORD)
  g1[1] = (int)(48u << 16);       // tensor_dim0 = 48
  g1[2] = (int)(48u << 16);       // tensor_dim1 = 48
  g1[3] = (int)(48u << 16);       // tile_dim0 = 48
  g1[4] = (int)(48u | (2u << 16));// tile_dim1 = 48, tile_dim2 = 2 channels
  g1[5] = 48;                     // tensor_dim0_stride = 48
  g1[6] = (int)(2304u << 16);     // tensor_dim1_stride[15:0] = 2304 (H*W)
  g1[7] = 0;                      // tensor_dim1_stride[47:16] = 0
  i32x4v g2; g2[0] = CCH; g2[1] = 0; g2[2] = 0; g2[3] = 0;  // tensor_dim2
  i32x4v g3; g3[0] = 0;   g3[1] = 0; g3[2] = 0; g3[3] = 0;
#if defined(__clang_major__) && (__clang_major__ >= 23)
  i32x8v gz; gz[0]=0; gz[1]=0; gz[2]=0; gz[3]=0; gz[4]=0; gz[5]=0; gz[6]=0; gz[7]=0;
  __builtin_amdgcn_tensor_load_to_lds(g0, g1, g2, g3, gz, 0);
#else
  __builtin_amdgcn_tensor_load_to_lds(g0, g1, g2, g3, 0);
#endif
}
#endif

__global__ __launch_bounds__(256)
void embKernel(const float* __restrict__ feat, float* dout, const char* __restrict__ ws) {
  const int cblk = blockIdx.x;          // 0..191
  const int n    = blockIdx.y;          // 0..3
  const int c0   = cblk * CB;
  const int tid  = threadIdx.x;
  __shared__ float tile[TILE_FLOATS];

  const float* gsrc = feat + ((size_t)n * CCH + c0) * (HF * HF);

#if __has_builtin(__builtin_amdgcn_tensor_load_to_lds)
  if (tid < 32) {  // one wave issues the TDM descriptors (EXEC ignored by TDM)
    tdm_load_2ch(gsrc,            &tile[0]);
    tdm_load_2ch(gsrc + 2 * 2304, &tile[DESC_FLOATS]);
#if __has_builtin(__builtin_amdgcn_s_wait_tensorcnt)
    __builtin_amdgcn_s_wait_tensorcnt(0);
#else
    asm volatile("s_wait_tensorcnt 0x0" ::: "memory");
#endif
  }
#else
  for (int k = tid; k < CB * 2304; k += 256) {
    int d = k / (2 * 2304);           // descriptor half
    int l = k - d * (2 * 2304);       // linear within half
    tile[d * DESC_FLOATS + l + (l >> 8)] = gsrc[k];
  }
#endif
  __syncthreads();

  const int p = tid & 63;    // point (fastest in emb layout -> coalesced stores)
  const int c = tid >> 6;    // 0..3 within channel block
  const int descBase = (c >> 1) * DESC_FLOATS;
  const int cpl      = (c & 1) * 2304;
  float* embBase = dout + 17408;

  for (int m = 0; m < MB; ++m) {
    const int nm = n * MB + m;
    int4   id = ((const int4*)(ws + WS_IDX))[nm * 64 + p];   // y0,x0,y1,x1
    float4 wv = ((const float4*)(ws + WS_WTS))[nm * 64 + p];
    float  rm = dout[1024 + nm * 64 + p];                     // roi_mask value
    int l00 = cpl + id.x * 48 + id.y;
    int l01 = cpl + id.x * 48 + id.w;
    int l10 = cpl + id.z * 48 + id.y;
    int l11 = cpl + id.z * 48 + id.w;
    float v00 = tile[descBase + l00 + (l00 >> 8)];
    float v01 = tile[descBase + l01 + (l01 >> 8)];
    float v10 = tile[descBase + l10 + (l10 >> 8)];
    float v11 = tile[descBase + l11 + (l11 >> 8)];
    float val = wv.x * v00 + wv.y * v01 + wv.z * v10 + wv.w * v11;
    __builtin_nontemporal_store(val * rm,
        &embBase[(size_t)nm * 49152 + (size_t)(c0 + c) * 64 + p]); // TH=NT store
  }
}

// ---------------------------------------------------------------------------
extern "C" void kernel_launch(void* const* d_in, const int* in_sizes, int n_in,
                              void* d_out, int out_size, void* d_ws, size_t ws_size,
                              hipStream_t stream) {
  const float* feat  = (const float*)d_in[0];   // (4,768,48,48) f32
  const float* masks = (const float*)d_in[1];   // (4,64,768,768) f32
  float* out = (float*)d_out;                   // [boxes 1024 | roi_masks 16384 | emb 12582912]
  char*  ws  = (char*)d_ws;

  hipLaunchKernelGGL(maskScanKernel, dim3(NM), dim3(1024), 0, stream, masks, out, ws);
  hipLaunchKernelGGL(pointKernel,    dim3(NM), dim3(64),   0, stream, out, ws);
  hipLaunchKernelGGL(embKernel,      dim3(192, NB), dim3(256), 0, stream, feat, out, ws);
}